// GraphRelationModule_31885837205812
// MI455X (gfx1250) — compile-verified
//
#include <hip/hip_runtime.h>
#include <hip/hip_bf16.h>

typedef float v2f __attribute__((ext_vector_type(2)));
typedef float v8f __attribute__((ext_vector_type(8)));

// ---------------------------------------------------------------------------
// GEMM: C[M,N] = act(A[M,K] @ W[K,N] + bias[N]); bias may be null.
// One 16(M) x 64(N) output tile per wave: 4 independent f32 WMMA accumulator
// chains sharing one A fragment per K-step (V_WMMA_F32_16X16X4_F32).
// Fragment layouts per CDNA5 ISA 7.12.2:
//   A 16x4:  lanes 0-15 -> M=lane, VGPR0=K0,VGPR1=K1 ; lanes 16-31 -> K2,K3
//   B 4x16:  VGPR0: lanes0-15 row K0, lanes16-31 row K2; VGPR1: K1 / K3
//   C 16x16: VGPR r: lanes0-15 (M=r,N=lane), lanes16-31 (M=r+8,N=lane-16)
// ---------------------------------------------------------------------------
__global__ __launch_bounds__(256) void wmma_gemm_bias(
    const float* __restrict__ A, const float* __restrict__ W,
    const float* __restrict__ bias, float* __restrict__ C,
    int M, int N, int K, int relu)
{
    const int lane = threadIdx.x & 31;
    const int wave = threadIdx.x >> 5;
    const int tilesN = N >> 6;                    // 64-wide N tiles
    const int tile = blockIdx.x * 8 + wave;
    const int totalTiles = (M >> 4) * tilesN;
    if (tile >= totalTiles) return;               // wave-uniform: EXEC all-1s
    const int tm = (tile / tilesN) << 4;
    const int tn = (tile % tilesN) << 6;

    const int half = lane >> 4;                   // 0: lanes 0-15, 1: 16-31
    const int l16  = lane & 15;

    const float* ap = A + (size_t)(tm + l16) * K + 2 * half;      // A[m, k+2h]
    const float* bp = W + (size_t)(2 * half) * N + tn + l16;      // W[k+2h, n]
    const size_t sN = (size_t)N;

    v8f acc0 = {}, acc1 = {}, acc2 = {}, acc3 = {};
    for (int k = 0; k < K; k += 4) {
        v2f a = *(const v2f*)ap;                  // A[m,k+2h], A[m,k+2h+1]
        const float* b0 = bp;                     // row k+2h
        const float* b1 = bp + sN;                // row k+2h+1
        v2f f0, f1, f2, f3;
        f0.x = b0[0];  f0.y = b1[0];              // immediate-offset b32 loads
        f1.x = b0[16]; f1.y = b1[16];
        f2.x = b0[32]; f2.y = b1[32];
        f3.x = b0[48]; f3.y = b1[48];
        acc0 = __builtin_amdgcn_wmma_f32_16x16x4_f32(false, a, false, f0,
                                                     (short)0, acc0, false, false);
        acc1 = __builtin_amdgcn_wmma_f32_16x16x4_f32(false, a, false, f1,
                                                     (short)0, acc1, false, false);
        acc2 = __builtin_amdgcn_wmma_f32_16x16x4_f32(false, a, false, f2,
                                                     (short)0, acc2, false, false);
        acc3 = __builtin_amdgcn_wmma_f32_16x16x4_f32(false, a, false, f3,
                                                     (short)0, acc3, false, false);
        ap += 4;
        bp += 4 * sN;
        __builtin_prefetch(bp + 8 * sN, 0, 1);    // global_prefetch_b8: next W panel
    }

    const int row0 = tm + half * 8;
    float bv0 = 0.f, bv1 = 0.f, bv2 = 0.f, bv3 = 0.f;
    if (bias) {
        const float* bb = bias + tn + l16;
        bv0 = bb[0]; bv1 = bb[16]; bv2 = bb[32]; bv3 = bb[48];
    }
    float* crow = C + (size_t)row0 * N + tn + l16;
#pragma unroll
    for (int r = 0; r < 8; ++r) {
        float v0 = acc0[r] + bv0, v1 = acc1[r] + bv1;
        float v2 = acc2[r] + bv2, v3 = acc3[r] + bv3;
        if (relu) {
            v0 = fmaxf(v0, 0.f); v1 = fmaxf(v1, 0.f);
            v2 = fmaxf(v2, 0.f); v3 = fmaxf(v3, 0.f);
        }
        crow[0] = v0; crow[16] = v1; crow[32] = v2; crow[48] = v3;
        crow += sN;
    }
}

// cnt[i] = #{j : y[j]==y[i], j!=i}
__global__ void count_same(const int* __restrict__ y, float* __restrict__ cnt, int S)
{
    int i = blockIdx.x * blockDim.x + threadIdx.x;
    if (i >= S) return;
    int yi = y[i];
    int c = 0;
    for (int j = 0; j < S; ++j) c += (y[j] == yi) ? 1 : 0;
    cnt[i] = (float)(c - 1);
}

// haggr[i,h] = sum_{j: y_j==y_i, j!=i} relu(A[i,h] + B[j,h] + bm1[h])
// (matmul with Wm2 distributes over this masked sum -> h tensor never built)
__global__ __launch_bounds__(256) void pair_accum(
    const float* __restrict__ Abuf, const float* __restrict__ Bbuf,
    const float* __restrict__ bm1t, const int* __restrict__ y,
    float* __restrict__ haggr, int S, int H)
{
    const int i = blockIdx.x;
    const int h = threadIdx.x;
    const int yi = y[i];
    const float av = Abuf[(size_t)i * H + h] + bm1t[h];
    float acc = 0.0f;
    for (int j = 0; j < S; ++j) {
        if (y[j] == yi && j != i)                 // block-uniform branch
            acc += fmaxf(av + Bbuf[(size_t)j * H + h], 0.0f);
    }
    haggr[(size_t)i * H + h] = acc;
}

// s[i,h] += (aggmat[i,h] + cnt[i]*bm2[h]) / max(cnt[i],1)
__global__ __launch_bounds__(256) void update_s(
    float* __restrict__ s, const float* __restrict__ aggmat,
    const float* __restrict__ bm2t, const float* __restrict__ cnt, int H)
{
    const int i = blockIdx.x;
    const int h = threadIdx.x;
    const float c = cnt[i];
    const float denom = fmaxf(c, 1.0f);
    s[(size_t)i * H + h] += (aggmat[(size_t)i * H + h] + c * bm2t[h]) / denom;
}

// scores[q,s] = sum_h relu(hq[q,h]+hs[s,h]+br1[h]) * wr2[h] + br2
// One block per q (8 waves); each wave owns one s at a time, lane covers 8 h.
__global__ __launch_bounds__(256) void score_kernel(
    const float* __restrict__ hq, const float* __restrict__ hs,
    const float* __restrict__ br1, const float* __restrict__ wr2,
    const float* __restrict__ br2, float* __restrict__ out, int S, int H)
{
    const int q = blockIdx.x;
    const int lane = threadIdx.x & 31;
    const int wave = threadIdx.x >> 5;
    const int hbase = lane * 8;                   // H == 256 == 32 lanes * 8

    float hqv[8], bv[8], wv[8];
#pragma unroll
    for (int u = 0; u < 8; ++u) {
        hqv[u] = hq[(size_t)q * H + hbase + u];
        bv[u]  = br1[hbase + u];
        wv[u]  = wr2[hbase + u];
    }
    const float b2 = br2[0];

    for (int s = wave; s < S; s += 8) {
        const float* hsrow = hs + (size_t)s * H + hbase;
        float sum = 0.0f;
#pragma unroll
        for (int u = 0; u < 8; ++u)
            sum += fmaxf(hqv[u] + hsrow[u] + bv[u], 0.0f) * wv[u];
#pragma unroll
        for (int off = 16; off > 0; off >>= 1)
            sum += __shfl_down(sum, off, 32);
        if (lane == 0) out[(size_t)q * S + s] = sum + b2;
    }
}

extern "C" void kernel_launch(void* const* d_in, const int* in_sizes, int n_in,
                              void* d_out, int out_size, void* d_ws, size_t ws_size,
                              hipStream_t stream)
{
    const float* Xq  = (const float*)d_in[0];   // (1024,512)
    const float* Xs  = (const float*)d_in[1];   // (512,512)
    const int*   y   = (const int*)d_in[2];     // (512,)
    const float* Wn1 = (const float*)d_in[3];   // (512,256)
    const float* bn1 = (const float*)d_in[4];
    const float* Wn2 = (const float*)d_in[5];   // (256,256)
    const float* bn2 = (const float*)d_in[6];
    const float* Wm1 = (const float*)d_in[7];   // (3,512,256)
    const float* bm1 = (const float*)d_in[8];   // (3,256)
    const float* Wm2 = (const float*)d_in[9];   // (3,256,256)
    const float* bm2 = (const float*)d_in[10];  // (3,256)
    const float* Wr1 = (const float*)d_in[11];  // (512,256)
    const float* br1 = (const float*)d_in[12];
    const float* wr2 = (const float*)d_in[13];  // (256,)
    const float* br2 = (const float*)d_in[14];  // scalar
    float* out = (float*)d_out;                 // (1024,512)

    const int E = 512, H = 256, Q = 1024, S = 512, STEPS = 3;

    float* ws  = (float*)d_ws;
    float* q1  = ws; ws += (size_t)Q * H;
    float* qf  = ws; ws += (size_t)Q * H;
    float* s1  = ws; ws += (size_t)S * H;
    float* sf  = ws; ws += (size_t)S * H;
    float* Ab  = ws; ws += (size_t)S * H;
    float* Bb  = ws; ws += (size_t)S * H;
    float* hag = ws; ws += (size_t)S * H;
    float* agg = ws; ws += (size_t)S * H;
    float* hqb = ws; ws += (size_t)Q * H;
    float* hsb = ws; ws += (size_t)S * H;
    float* cnt = ws; ws += S;

    auto gemm = [&](const float* A, const float* W, const float* bias, float* C,
                    int M, int N, int K, int relu) {
        int tiles  = (M / 16) * (N / 64);
        int blocks = (tiles + 7) / 8;
        wmma_gemm_bias<<<blocks, 256, 0, stream>>>(A, W, bias, C, M, N, K, relu);
    };

    // Feature encoders (shared MLP)
    gemm(Xq, Wn1, bn1, q1, Q, H, E, 1);
    gemm(q1, Wn2, bn2, qf, Q, H, H, 0);
    gemm(Xs, Wn1, bn1, s1, S, H, E, 1);
    gemm(s1, Wn2, bn2, sf, S, H, H, 0);

    count_same<<<(S + 255) / 256, 256, 0, stream>>>(y, cnt, S);

    // Message-passing steps (msg GEMM factored out of the masked pair sum)
    for (int t = 0; t < STEPS; ++t) {
        const float* W1a = Wm1 + (size_t)t * (2 * H) * H;
        const float* W1b = W1a + (size_t)H * H;
        gemm(sf, W1a, nullptr, Ab, S, H, H, 0);
        gemm(sf, W1b, nullptr, Bb, S, H, H, 0);
        pair_accum<<<S, H, 0, stream>>>(Ab, Bb, bm1 + (size_t)t * H, y, hag, S, H);
        gemm(hag, Wm2 + (size_t)t * H * H, nullptr, agg, S, H, H, 0);
        update_s<<<S, H, 0, stream>>>(sf, agg, bm2 + (size_t)t * H, cnt, H);
    }

    // Relation head
    gemm(qf, Wr1, nullptr, hqb, Q, H, H, 0);
    gemm(sf, Wr1 + (size_t)H * H, nullptr, hsb, S, H, H, 0);
    score_kernel<<<Q, 256, 0, stream>>>(hqb, hsb, br1, wr2, br2, out, S, H);
}